// SparseMedianInterpolator_26207890440545
// MI455X (gfx1250) — compile-verified
//
#include <hip/hip_runtime.h>
#include <cstdint>
#include <cstddef>

// Problem constants (reference: shape (4,1,512,640), K=5, prior=0)
#define W_   640
#define H_   512
#define NIMG 4
#define TW   64          // tile width  (outputs)
#define TH   32          // tile height (outputs)
#define PADR 2           // (K-1)/2
#define LW   (TW + 4)    // 68 padded tile width
#define LH   (TH + 4)    // 36 padded tile height
#define NTHREADS 256     // 8 wave32s

__device__ __forceinline__ void ce(float& a, float& b) {
    float lo = fminf(a, b);
    float hi = fmaxf(a, b);
    a = lo; b = hi;
}

__device__ __forceinline__ int reflect_idx(int i, int n) {
    i = (i < 0) ? -i : i;                 // numpy/jax 'reflect': -1 -> 1, -2 -> 2
    i = (i >= n) ? (2 * n - 2 - i) : i;   // n -> n-2, n+1 -> n-3
    return i;
}

__global__ __launch_bounds__(NTHREADS)
void median5_nan_interp_kernel(const float* __restrict__ x, float* __restrict__ out) {
    __shared__ float smem[LH * LW];

    const int img = blockIdx.z;
    const float* src = x + (size_t)img * ((size_t)H_ * W_);
    const int gx0 = blockIdx.x * TW - PADR;
    const int gy0 = blockIdx.y * TH - PADR;

    // ---- Stage padded tile into LDS via CDNA5 async global->LDS copies ----
    for (int idx = threadIdx.x; idx < LH * LW; idx += NTHREADS) {
        const int py = idx / LW;
        const int px = idx - py * LW;
        const int gy = reflect_idx(gy0 + py, H_);
        const int gx = reflect_idx(gx0 + px, W_);
        const float* gp = src + (size_t)gy * W_ + gx;
        // AS3 offset == low 32 bits of the flat LDS-aperture address
        unsigned lds_off = (unsigned)(unsigned long long)&smem[idx];
        asm volatile("global_load_async_to_lds_b32 %0, %1, off"
                     :: "v"(lds_off), "v"(gp)
                     : "memory");
    }
    asm volatile("s_wait_asynccnt 0x0" ::: "memory");
    __syncthreads();

    // ---- Compute: each thread owns a 1x8 vertical output strip ----
    const int tx  = threadIdx.x & (TW - 1);   // 0..63
    const int tyq = threadIdx.x >> 6;         // 0..3
    const int y0p = tyq * 8;                  // first padded row of this strip

    float srow[5][5];   // ring of HORIZONTALLY SORTED sanitized window rows
    float centv[5];     // sanitized center value of each ring row
    bool  nanc[5];      // was the center value NaN?

    auto load_row = [&](int slot, int prow) {
        const float* rp = &smem[prow * LW + tx];
        float v[5];
        #pragma unroll
        for (int j = 0; j < 5; ++j) {
            const float raw = rp[j];
            const bool ok = (raw == raw);
            v[j] = ok ? raw : 0.0f;
            if (j == 2) { nanc[slot] = !ok; centv[slot] = v[j]; }
        }
        // Knuth optimal 9-CE 5-sorter
        ce(v[0], v[1]); ce(v[3], v[4]); ce(v[2], v[4]);
        ce(v[2], v[3]); ce(v[0], v[3]); ce(v[0], v[2]);
        ce(v[1], v[4]); ce(v[1], v[3]); ce(v[1], v[2]);
        #pragma unroll
        for (int j = 0; j < 5; ++j) srow[slot][j] = v[j];
    };

    #pragma unroll
    for (int r = 0; r < 4; ++r) load_row(r, y0p + r);

    float* dst = out + (size_t)img * ((size_t)H_ * W_);

    #pragma unroll
    for (int o = 0; o < 8; ++o) {
        load_row((o + 4) % 5, y0p + o + 4);

        // Window = the 5 ring rows (any row order is fine: rows stay sorted).
        float w[25];
        #pragma unroll
        for (int i = 0; i < 5; ++i)
            #pragma unroll
            for (int j = 0; j < 5; ++j) w[i * 5 + j] = srow[i][j];

        // Sort each column (columns of a row-sorted matrix -> both sorted).
        #pragma unroll
        for (int j = 0; j < 5; ++j) {
            ce(w[0 + j],  w[5 + j]);  ce(w[15 + j], w[20 + j]);
            ce(w[10 + j], w[20 + j]); ce(w[10 + j], w[15 + j]);
            ce(w[0 + j],  w[15 + j]); ce(w[0 + j],  w[10 + j]);
            ce(w[5 + j],  w[20 + j]); ce(w[5 + j],  w[15 + j]);
            ce(w[5 + j],  w[10 + j]);
        }

        // Rank-13 of 25 must be one of 13 candidates forming 5 sorted chains:
        // A=[w3,w4] B=[w7,w8,w9] C=[w11,w12,w13] D=[w15,w16,w17] E=[w20,w21]
        // (6 elements are provably below rank 13, 6 above.)

        // AE = odd-even merge(2,2): sorted [A0,A1,E0,E1]
        float A0 = w[3],  A1 = w[4],  E0 = w[20], E1 = w[21];
        ce(A0, E0); ce(A1, E1); ce(A1, E0);

        // V = odd-even merge(3,3) of B and D: sorted [B0,B1,B2,D1,D0,D2]
        float B0 = w[7],  B1 = w[8],  B2 = w[9];
        float D0 = w[15], D1 = w[16], D2 = w[17];
        ce(B0, D0); ce(B2, D2); ce(B2, D0);   // O = [B0,B2,D0,D2]
        ce(B1, D1);                            // E = [B1,D1]
        ce(B1, B2); ce(D1, D0);                // fixups
        // V sorted = B0, B1, B2, D1, D0, D2

        // U = odd-even merge(4,3) of AE=[A0,A1,E0,E1] and C=[C0,C1,C2]
        float C0 = w[11], C1 = w[12], C2 = w[13];
        ce(A0, C0); ce(E0, C2); ce(E0, C0);   // O = merge([A0,E0],[C0,C2])
        ce(A1, C1); ce(E1, C1);               // E = merge([A1,E1],[C1])
        ce(A1, E0); ce(E1, C0); ce(C1, C2);   // fixups
        // U sorted = A0, A1, E0, E1, C0, C1, C2

        // 7th smallest of U(7) ∪ V(6):  min_i max(U[i], V[6-i-? ])
        float med = fmaxf(A0, D2);                 // max(U0, V5)
        med = fminf(med, fmaxf(A1, D0));           // max(U1, V4)
        med = fminf(med, fmaxf(E0, D1));           // max(U2, V3)
        med = fminf(med, fmaxf(E1, B2));           // max(U3, V2)
        med = fminf(med, fmaxf(C0, B1));           // max(U4, V1)
        med = fminf(med, fmaxf(C1, B0));           // max(U5, V0)
        med = fminf(med, C2);                      // U6

        const int cs = (o + 2) % 5;                 // center row slot
        const float res = nanc[cs] ? med : centv[cs];

        const int gy = blockIdx.y * TH + y0p + o;
        const int gx = blockIdx.x * TW + tx;
        dst[(size_t)gy * W_ + gx] = res;
    }
}

extern "C" void kernel_launch(void* const* d_in, const int* in_sizes, int n_in,
                              void* d_out, int out_size, void* d_ws, size_t ws_size,
                              hipStream_t stream) {
    (void)in_sizes; (void)n_in; (void)out_size; (void)d_ws; (void)ws_size;
    const float* x = (const float*)d_in[0];
    float* out = (float*)d_out;
    dim3 grid(W_ / TW, H_ / TH, NIMG);   // 10 x 16 x 4 tiles, exact fit
    dim3 block(NTHREADS);
    median5_nan_interp_kernel<<<grid, block, 0, stream>>>(x, out);
}